// LightGCNLayer_87866440942260
// MI455X (gfx1250) — compile-verified
//
#include <hip/hip_runtime.h>

// LightGCN propagation layer for MI455X (gfx1250, wave32).
//
// updated_users = segment_sum(edge_val * item_emb[edge_item], edge_user)  [200000 x 64]
// updated_items = segment_sum(edge_val * user_emb[edge_user], edge_item)  [100000 x 64]
//
// Roofline: 0.82 GFLOP vs ~192 MB working set (fits MI455X's 192 MB L2).
// Pure gather/scatter, AI ~0.25 FLOP/B -> bounded by L2 gather + atomic
// throughput, not math. No dense contraction exists, so no WMMA.
//
// Design:
//  - one half-wave (16 lanes) per edge; each lane owns 4 of 64 dims
//    -> every row gather is a global_load_b128 (16 lanes x 16B = 256B row)
//  - scatter-add via no-return global_atomic_add_f32 at device scope
//  - edge lists (38.4 MB, touched once) loaded non-temporal so they don't
//    evict the L2-resident embeddings/outputs
//  - register-rotated software pipeline + global_prefetch_b8 on the next
//    iteration's random rows to hide gather latency

#define EMBED_DIM 64

__global__ void zero_out_kernel(float4* __restrict__ out4, long n4,
                                float* __restrict__ out_tail, int ntail) {
    long i = (long)blockIdx.x * blockDim.x + threadIdx.x;
    long stride = (long)gridDim.x * blockDim.x;
    float4 z = make_float4(0.f, 0.f, 0.f, 0.f);
    for (; i < n4; i += stride) out4[i] = z;           // global_store_b128
    if (blockIdx.x == 0 && threadIdx.x < (unsigned)ntail) out_tail[threadIdx.x] = 0.f;
}

// No-return hardware float atomic, explicitly device scope (RMW performed at
// the device coherence point; avoids any compiler CAS-loop fallback).
__device__ __forceinline__ void gatomic_add_f32(float* p, float v) {
    asm volatile("global_atomic_add_f32 %0, %1, off scope:SCOPE_DEV"
                 :: "v"(p), "v"(v) : "memory");
}

__global__ void __launch_bounds__(256)
lightgcn_scatter_kernel(const float* __restrict__ user_emb,
                        const float* __restrict__ item_emb,
                        const int*   __restrict__ edge_user,
                        const int*   __restrict__ edge_item,
                        const float* __restrict__ edge_val,
                        float* __restrict__ out_users,
                        float* __restrict__ out_items,
                        int nedges) {
    const int lane = threadIdx.x & 31;      // wave32
    const int half = lane >> 4;             // half-wave id: 0 / 1
    const int sub  = lane & 15;             // lane within half-wave
    const int d    = sub << 2;              // this lane's 4 dims [d, d+3]

    const int gthread  = blockIdx.x * blockDim.x + threadIdx.x;
    const int nhalf    = ((gridDim.x * blockDim.x) >> 5) << 1;
    int e = ((gthread >> 5) << 1) + half;   // 2 edges per wave per iteration
    if (e >= nedges) return;

    // Prologue of the rotated pipeline: non-temporal streaming loads (edge
    // lists are single-use; keep them out of the reused L2 footprint).
    int   u  = __builtin_nontemporal_load(edge_user + e);
    int   it = __builtin_nontemporal_load(edge_item + e);
    float v  = __builtin_nontemporal_load(edge_val  + e);

    for (;;) {
        const int  en       = e + nhalf;
        const bool has_next = en < nedges;
        int un = 0, itn = 0;
        float vn = 0.f;
        if (has_next) {
            un  = __builtin_nontemporal_load(edge_user + en);
            itn = __builtin_nontemporal_load(edge_item + en);
            vn  = __builtin_nontemporal_load(edge_val  + en);
            // Prefetch next iteration's random embedding rows (global_prefetch_b8).
            __builtin_prefetch(&user_emb[(long)un  * EMBED_DIM + d], 0, 3);
            __builtin_prefetch(&item_emb[(long)itn * EMBED_DIM + d], 0, 3);
        }

        // 128-bit row gathers: half-wave pulls one full 256B row per matrix.
        const float4 irow = *(const float4*)(item_emb + (long)it * EMBED_DIM + d);
        const float4 urow = *(const float4*)(user_emb + (long)u  * EMBED_DIM + d);

        float* up = out_users + (long)u  * EMBED_DIM + d;
        float* ip = out_items + (long)it * EMBED_DIM + d;

        // Scatter-add via native L2 f32 atomics (no return -> STOREcnt path).
        gatomic_add_f32(up + 0, v * irow.x);
        gatomic_add_f32(up + 1, v * irow.y);
        gatomic_add_f32(up + 2, v * irow.z);
        gatomic_add_f32(up + 3, v * irow.w);
        gatomic_add_f32(ip + 0, v * urow.x);
        gatomic_add_f32(ip + 1, v * urow.y);
        gatomic_add_f32(ip + 2, v * urow.z);
        gatomic_add_f32(ip + 3, v * urow.w);

        if (!has_next) break;
        e = en; u = un; it = itn; v = vn;   // rotate pipeline registers
    }
}

extern "C" void kernel_launch(void* const* d_in, const int* in_sizes, int n_in,
                              void* d_out, int out_size, void* d_ws, size_t ws_size,
                              hipStream_t stream) {
    const float* user_emb = (const float*)d_in[0];   // [NUM_USERS * 64] f32
    const float* item_emb = (const float*)d_in[1];   // [NUM_ITEMS * 64] f32
    const int*   edge_u   = (const int*)  d_in[2];   // [E] i32
    const int*   edge_i   = (const int*)  d_in[3];   // [E] i32
    const float* edge_v   = (const float*)d_in[4];   // [E] f32

    const int nusers = in_sizes[0] / EMBED_DIM;
    const int nedges = in_sizes[2];

    float* out_users = (float*)d_out;
    float* out_items = out_users + (long)nusers * EMBED_DIM;

    // 1) Zero the (poisoned) output — the scatter pass accumulates into it.
    const long n4    = (long)out_size >> 2;
    const int  ntail = out_size & 3;
    int zblocks = (int)((n4 + 255) / 256);
    if (zblocks > 32768) zblocks = 32768;
    if (zblocks < 1)     zblocks = 1;
    zero_out_kernel<<<zblocks, 256, 0, stream>>>((float4*)d_out, n4,
                                                 (float*)d_out + (n4 << 2), ntail);

    // 2) Edge-parallel gather + scatter-add. 4096 blocks x 256 thr = 32768
    //    waves = 65536 half-waves; ~49 grid-stride iterations over 3.2M edges.
    lightgcn_scatter_kernel<<<4096, 256, 0, stream>>>(
        user_emb, item_emb, edge_u, edge_i, edge_v,
        out_users, out_items, nedges);
}